// GaussianModel_52398601011249
// MI455X (gfx1250) — compile-verified
//
#include <hip/hip_runtime.h>
#include <hip/hip_bf16.h>

#define N_GAUSS 4096
#define RES     64
#define NB      16
#define SPLIT   4
#define KPAD    12   // 10 features padded to 12 (3 x K=4 WMMA slices)
#define STAGE_G 64   // gaussians staged into LDS per async stage (3 KB)

// Workspace layout (floats):
//  [0..3]              : center.xyz, scale
//  [OFF_OPA  .. +4096) : culled opacities
//  [OFF_FEAT .. +4096*12) : per-gaussian 12-dim feature rows
//  [OFF_XYZ  .. +4096*4)  : scaled gaussian centers (float4)
#define OFF_OPA  16
#define OFF_FEAT (OFF_OPA + N_GAUSS)
#define OFF_XYZ  (OFF_FEAT + N_GAUSS * KPAD)
// total ~279 KB of d_ws

typedef float v2f __attribute__((ext_vector_type(2)));
typedef float v8f __attribute__((ext_vector_type(8)));

// --- CDNA5 async global->LDS copy (ASYNCcnt-tracked), inline asm for
// portability across ROCm7.2 / amdgpu-toolchain builtin arity differences.
__device__ __forceinline__ void async_load_b64_to_lds(unsigned lds_byte_off,
                                                      const float* gptr) {
    asm volatile("global_load_async_to_lds_b64 %0, %1, off"
                 :: "v"(lds_byte_off), "v"(gptr)
                 : "memory");
}
__device__ __forceinline__ void wait_asynccnt0() {
    asm volatile("s_wait_asynccnt 0x0" ::: "memory");
}

// ---------------------------------------------------------------------------
// Kernel 0: sigmoid/cull opacities, global min/max of kept xyz -> center/scale
// ---------------------------------------------------------------------------
__global__ __launch_bounds__(256)
void k_reduce(const float* __restrict__ xyz, const float* __restrict__ opacity,
              float* __restrict__ W) {
    __shared__ float s[6][256];
    const int tid = threadIdx.x;
    const float BIG = 1e10f;
    float mnx = BIG, mny = BIG, mnz = BIG;
    float mxx = -BIG, mxy = -BIG, mxz = -BIG;
    for (int g = tid; g < N_GAUSS; g += 256) {
        float o   = opacity[g];
        float sg  = 1.0f / (1.0f + __expf(-o));
        bool keep = sg > 0.005f;
        W[OFF_OPA + g] = keep ? sg : 0.0f;
        float x = xyz[3 * g + 0], y = xyz[3 * g + 1], z = xyz[3 * g + 2];
        mnx = fminf(mnx, keep ? x : BIG);
        mny = fminf(mny, keep ? y : BIG);
        mnz = fminf(mnz, keep ? z : BIG);
        mxx = fmaxf(mxx, keep ? x : -BIG);
        mxy = fmaxf(mxy, keep ? y : -BIG);
        mxz = fmaxf(mxz, keep ? z : -BIG);
    }
    s[0][tid] = mnx; s[1][tid] = mny; s[2][tid] = mnz;
    s[3][tid] = mxx; s[4][tid] = mxy; s[5][tid] = mxz;
    __syncthreads();
    for (int stride = 128; stride > 0; stride >>= 1) {
        if (tid < stride) {
            s[0][tid] = fminf(s[0][tid], s[0][tid + stride]);
            s[1][tid] = fminf(s[1][tid], s[1][tid + stride]);
            s[2][tid] = fminf(s[2][tid], s[2][tid + stride]);
            s[3][tid] = fmaxf(s[3][tid], s[3][tid + stride]);
            s[4][tid] = fmaxf(s[4][tid], s[4][tid + stride]);
            s[5][tid] = fmaxf(s[5][tid], s[5][tid + stride]);
        }
        __syncthreads();
    }
    if (tid == 0) {
        float dx = s[3][0] - s[0][0];
        float dy = s[4][0] - s[1][0];
        float dz = s[5][0] - s[2][0];
        float scale = 1.8f / fmaxf(dx, fmaxf(dy, dz));
        W[0] = 0.5f * (s[0][0] + s[3][0]);
        W[1] = 0.5f * (s[1][0] + s[4][0]);
        W[2] = 0.5f * (s[2][0] + s[5][0]);
        W[3] = scale;
    }
}

// ---------------------------------------------------------------------------
// Kernel 1: per-gaussian quadratic-form features (12-dim) + scaled centers
// ---------------------------------------------------------------------------
__global__ __launch_bounds__(256)
void k_feat(const float* __restrict__ xyz, const float* __restrict__ scaling,
            const float* __restrict__ rot, float* __restrict__ W) {
    const int g = blockIdx.x * 256 + threadIdx.x;
    if (g >= N_GAUSS) return;
    const float cx = W[0], cy = W[1], cz = W[2], sc = W[3];

    const float gx = (xyz[3 * g + 0] - cx) * sc;
    const float gy = (xyz[3 * g + 1] - cy) * sc;
    const float gz = (xyz[3 * g + 2] - cz) * sc;

    const float s0 = __expf(scaling[3 * g + 0]) * sc;
    const float s1 = __expf(scaling[3 * g + 1]) * sc;
    const float s2 = __expf(scaling[3 * g + 2]) * sc;

    float qr = rot[4 * g + 0], qx = rot[4 * g + 1];
    float qy = rot[4 * g + 2], qz = rot[4 * g + 3];
    const float qn = rsqrtf(qr * qr + qx * qx + qy * qy + qz * qz);
    qr *= qn; qx *= qn; qy *= qn; qz *= qn;

    float R[3][3];
    R[0][0] = 1.0f - 2.0f * (qy * qy + qz * qz);
    R[0][1] = 2.0f * (qx * qy - qr * qz);
    R[0][2] = 2.0f * (qx * qz + qr * qy);
    R[1][0] = 2.0f * (qx * qy + qr * qz);
    R[1][1] = 1.0f - 2.0f * (qx * qx + qz * qz);
    R[1][2] = 2.0f * (qy * qz - qr * qx);
    R[2][0] = 2.0f * (qx * qz - qr * qy);
    R[2][1] = 2.0f * (qy * qz + qr * qx);
    R[2][2] = 1.0f - 2.0f * (qx * qx + qy * qy);

    float L[3][3];
    #pragma unroll
    for (int i = 0; i < 3; ++i) {
        L[i][0] = R[i][0] * s0; L[i][1] = R[i][1] * s1; L[i][2] = R[i][2] * s2;
    }
    // C = L * L^T (symmetric)
    const float a = L[0][0]*L[0][0] + L[0][1]*L[0][1] + L[0][2]*L[0][2];
    const float b = L[0][0]*L[1][0] + L[0][1]*L[1][1] + L[0][2]*L[1][2];
    const float c = L[0][0]*L[2][0] + L[0][1]*L[2][1] + L[0][2]*L[2][2];
    const float d = L[1][0]*L[1][0] + L[1][1]*L[1][1] + L[1][2]*L[1][2];
    const float e = L[1][0]*L[2][0] + L[1][1]*L[2][1] + L[1][2]*L[2][2];
    const float f = L[2][0]*L[2][0] + L[2][1]*L[2][1] + L[2][2]*L[2][2];

    const float inv_det = 1.0f / (a*d*f + 2.0f*e*c*b - e*e*a - c*c*d - b*b*f + 1e-24f);
    const float ia = (d*f - e*e) * inv_det;
    const float ib = (e*c - b*f) * inv_det;
    const float ic = (e*b - c*d) * inv_det;
    const float id = (a*f - c*c) * inv_det;
    const float ie = (b*c - e*a) * inv_det;
    const float iff = (a*d - b*b) * inv_det;

    // power = A px^2 + D py^2 + F pz^2 + B px py + C px pz + E py pz
    const float A = -0.5f * ia, D = -0.5f * id, F = -0.5f * iff;
    const float B = -ib, C = -ic, E = -ie;
    // expand around gaussian center -> 10-dim feature (pad to 12)
    float* fr = W + OFF_FEAT + g * KPAD;
    fr[0] = A; fr[1] = D; fr[2] = F;
    fr[3] = B; fr[4] = C; fr[5] = E;
    fr[6] = -2.0f * A * gx - B * gy - C * gz;
    fr[7] = -2.0f * D * gy - B * gx - E * gz;
    fr[8] = -2.0f * F * gz - C * gx - E * gy;
    fr[9] = A*gx*gx + D*gy*gy + F*gz*gz + B*gx*gy + C*gx*gz + E*gy*gz;
    fr[10] = 0.0f; fr[11] = 0.0f;

    float4* xs4 = (float4*)(W + OFF_XYZ);
    xs4[g] = make_float4(gx, gy, gz, 0.0f);
}

// ---------------------------------------------------------------------------
// Kernel 2: per 4^3 block, power = pointFeat(64x12) @ gaussFeat^T(12x4096)
// via V_WMMA_F32_16X16X4_F32. Gaussian features are staged into LDS with
// double-buffered GLOBAL_LOAD_ASYNC_TO_LDS_B64 (ASYNCcnt), shared by all
// 4 waves; val = sum_g w_g * exp(power<=0 ? power : -inf).
// ---------------------------------------------------------------------------
__global__ __launch_bounds__(128)
void k_field(const float* __restrict__ W, float* __restrict__ out) {
    __shared__ float s_w[N_GAUSS];                          // 16 KB
    __shared__ __align__(16) float s_feat[2][STAGE_G * KPAD]; // 2 x 3 KB

    const int bid = blockIdx.x;
    const int bk = bid & 15, bj = (bid >> 4) & 15, bi = bid >> 8;
    const float step  = 2.0f / 63.0f;
    const float relax = (2.0f / NB) * 1.5f;   // block_size * relax_ratio

    const float vminx = (-1.0f + (bi * 4) * step) - relax;
    const float vmaxx = (-1.0f + (bi * 4 + 3) * step) + relax;
    const float vminy = (-1.0f + (bj * 4) * step) - relax;
    const float vmaxy = (-1.0f + (bj * 4 + 3) * step) + relax;
    const float vminz = (-1.0f + (bk * 4) * step) - relax;
    const float vmaxz = (-1.0f + (bk * 4 + 3) * step) + relax;

    const float*  opa  = W + OFF_OPA;
    const float*  feat = W + OFF_FEAT;
    const float4* xs4  = (const float4*)(W + OFF_XYZ);

    const int tid = threadIdx.x;
    for (int g = tid; g < N_GAUSS; g += 128) {
        float4 p = xs4[g];
        bool inside = (p.x > vminx) && (p.x < vmaxx) &&
                      (p.y > vminy) && (p.y < vmaxy) &&
                      (p.z > vminz) && (p.z < vmaxz);
        s_w[g] = inside ? opa[g] : 0.0f;
    }

    // LDS byte offsets of the two staging buffers (truncation of the flat
    // LDS-aperture address yields the in-allocation byte offset).
    const unsigned feat_lds0 = (unsigned)(uintptr_t)(&s_feat[0][0]);
    const unsigned buf_bytes = STAGE_G * KPAD * 4;

    // issue async stage: 64 rows x 12 floats = 384 b64 elements, 3 per thread
    auto issue_stage = [&](int buf, int n0) {
        const float* src = feat + n0 * KPAD;
        const unsigned dst = feat_lds0 + (unsigned)buf * buf_bytes;
        #pragma unroll
        for (int k = 0; k < 3; ++k) {
            const int e = tid + k * 128;            // b64 element 0..383
            async_load_b64_to_lds(dst + (unsigned)e * 8u, src + e * 2);
        }
    };

    issue_stage(0, 0);
    wait_asynccnt0();
    __syncthreads();        // s_w ready + stage 0 resident in LDS

    const int lane = tid & 31;
    const int wave = tid >> 5;
    const int m    = lane & 15;        // matrix row (A) / column (B,D)
    const int ksel = (lane >= 16) ? 2 : 0;

    // this wave's 16 points: x fixed per wave, (y,z) from row index m
    const float X = -1.0f + (bi * 4 + wave) * step;
    const float Y = -1.0f + (bj * 4 + (m >> 2)) * step;
    const float Z = -1.0f + (bk * 4 + (m & 3)) * step;

    float pf[KPAD] = { X*X, Y*Y, Z*Z, X*Y, X*Z, Y*Z, X, Y, Z, 1.0f, 0.0f, 0.0f };
    const v2f a0 = { pf[0 + ksel], pf[1 + ksel] };
    const v2f a1 = { pf[4 + ksel], pf[5 + ksel] };
    const v2f a2 = { pf[8 + ksel], pf[9 + ksel] };

    float acc[8] = {0.f, 0.f, 0.f, 0.f, 0.f, 0.f, 0.f, 0.f};

    const int NSTAGE = N_GAUSS / STAGE_G;   // 64
    for (int stage = 0; stage < NSTAGE; ++stage) {
        const int buf = stage & 1;
        if (stage + 1 < NSTAGE) issue_stage(buf ^ 1, (stage + 1) * STAGE_G);

        const float* fbuf = &s_feat[buf][0];
        #pragma unroll
        for (int cc = 0; cc < STAGE_G / 16; ++cc) {
            const int row = cc * 16 + m;                 // gaussian within stage
            const float* fb = fbuf + row * KPAD + ksel;  // LDS -> ds_load_b64
            const v2f b0 = *(const v2f*)(fb);
            const v2f b1 = *(const v2f*)(fb + 4);
            const v2f b2 = *(const v2f*)(fb + 8);
            const float w = s_w[stage * STAGE_G + cc * 16 + m];

            v8f cacc = {0.f, 0.f, 0.f, 0.f, 0.f, 0.f, 0.f, 0.f};
            cacc = __builtin_amdgcn_wmma_f32_16x16x4_f32(false, a0, false, b0,
                                                         (short)0, cacc, false, false);
            cacc = __builtin_amdgcn_wmma_f32_16x16x4_f32(false, a1, false, b1,
                                                         (short)0, cacc, false, false);
            cacc = __builtin_amdgcn_wmma_f32_16x16x4_f32(false, a2, false, b2,
                                                         (short)0, cacc, false, false);
            #pragma unroll
            for (int r = 0; r < 8; ++r) {
                float pw = cacc[r];
                float ev = (pw > 0.0f) ? 0.0f : __expf(pw);  // power>0 -> exp==0
                acc[r] += w * ev;
            }
        }
        wait_asynccnt0();   // next buffer landed (this wave's issues)
        __syncthreads();    // ... and everyone's; also fences buffer reuse
    }

    // sum over the 16 gaussian columns (lanes 0-15 / 16-31 hold disjoint rows)
    #pragma unroll
    for (int r = 0; r < 8; ++r) {
        float v = acc[r];
        v += __shfl_xor(v, 1, 32);
        v += __shfl_xor(v, 2, 32);
        v += __shfl_xor(v, 4, 32);
        v += __shfl_xor(v, 8, 32);
        acc[r] = v;
    }

    // D layout: VGPR r holds row M=r (lanes 0-15) / M=8+r (lanes 16-31)
    if ((lane & 15) == 0) {
        const int rowbase = (lane >= 16) ? 8 : 0;
        #pragma unroll
        for (int r = 0; r < 8; ++r) {
            const int mm = rowbase + r;
            const int oi = (bi * 4 + wave) * (RES * RES) +
                           (bj * 4 + (mm >> 2)) * RES +
                           (bk * 4 + (mm & 3));
            out[oi] = acc[r];
        }
    }
}

// ---------------------------------------------------------------------------
extern "C" void kernel_launch(void* const* d_in, const int* in_sizes, int n_in,
                              void* d_out, int out_size, void* d_ws, size_t ws_size,
                              hipStream_t stream) {
    const float* xyz     = (const float*)d_in[0];
    const float* scaling = (const float*)d_in[1];
    const float* rot     = (const float*)d_in[2];
    const float* opac    = (const float*)d_in[3];
    // d_in[4] = resolution (64), d_in[5] = num_blocks (16): baked in as constants
    float* W = (float*)d_ws;   // needs ~280 KB

    k_reduce<<<1, 256, 0, stream>>>(xyz, opac, W);
    k_feat<<<(N_GAUSS + 255) / 256, 256, 0, stream>>>(xyz, scaling, rot, W);
    k_field<<<NB * NB * NB, 128, 0, stream>>>(W, (float*)d_out);
}